// MLTModel_9174050144660
// MI455X (gfx1250) — compile-verified
//
#include <hip/hip_runtime.h>

typedef float v2f __attribute__((ext_vector_type(2)));
typedef float v8f __attribute__((ext_vector_type(8)));

#define NF 64
#define ROWS_PER_BLOCK 2048
#define THREADS 256
#define KMAX 64

__device__ __forceinline__ v8f wmma_f32_k4(v2f a, v2f b, v8f c) {
  // D = A(16x4 f32) * B(4x16 f32) + C(16x16 f32), exact fp32 accumulate
  return __builtin_amdgcn_wmma_f32_16x16x4_f32(
      /*neg_a=*/false, a, /*neg_b=*/false, b,
      /*c_mod=*/(short)0, c, /*reuse_a=*/false, /*reuse_b=*/false);
}

// Phase 1: stream 2048 rows/block, WMMA-GEMV into LDS scores, block top-k -> ws
__global__ void __launch_bounds__(THREADS)
gemv_topk_block(const float* __restrict__ A, const int* __restrict__ item_p,
                const int* __restrict__ size_p, float* __restrict__ ws_s,
                int* __restrict__ ws_i, int n_items) {
  __shared__ float u_lds[NF];
  __shared__ float sc[ROWS_PER_BLOCK];
  __shared__ float red_s[THREADS];
  __shared__ int   red_i[THREADS];

  const int tid  = threadIdx.x;
  const int lane = tid & 31;          // wave32
  const int wave = tid >> 5;          // 8 waves per block
  const int blockStart = blockIdx.x * ROWS_PER_BLOCK;
  const float NEG_INF = -__builtin_inff();

  if (tid < NF) u_lds[tid] = A[(size_t)item_p[0] * NF + tid];
  __syncthreads();

  // B operand (broadcast u): lanes 0-15 supply K={4s,4s+1}, lanes 16-31 K={4s+2,4s+3}
  const int koff = (lane >> 4) * 2;
  v2f bv[16];
#pragma unroll
  for (int s = 0; s < 16; ++s) {
    bv[s].x = u_lds[s * 4 + koff];
    bv[s].y = u_lds[s * 4 + koff + 1];
  }

  // Each wave: 8 steps x (two 16-row WMMA tiles) = 256 rows
#pragma unroll 1
  for (int it = 0; it < 8; ++it) {
    const int tileLocal = wave * 256 + it * 32;
    const int row0 = blockStart + tileLocal + (lane & 15);
    const int row1 = row0 + 16;
    const int r0c = row0 < n_items ? row0 : (n_items - 1);
    const int r1c = row1 < n_items ? row1 : (n_items - 1);
    const float* p0 = A + (size_t)r0c * NF + koff;
    const float* p1 = A + (size_t)r1c * NF + koff;
    v8f c0 = {};
    v8f c1 = {};
#pragma unroll
    for (int s = 0; s < 16; ++s) {
      v2f a0 = *(const v2f*)(p0 + s * 4);
      c0 = wmma_f32_k4(a0, bv[s], c0);
    }
#pragma unroll
    for (int s = 0; s < 16; ++s) {
      v2f a1 = *(const v2f*)(p1 + s * 4);
      c1 = wmma_f32_k4(a1, bv[s], c1);
    }
    // All D columns identical; lane0 holds M=0..7 in c[0..7], lane16 holds M=8..15
    if ((lane & 15) == 0) {
      const int half8 = (lane >> 4) * 8;
#pragma unroll
      for (int i = 0; i < 8; ++i) {
        sc[tileLocal + half8 + i]      = c0[i];
        sc[tileLocal + 16 + half8 + i] = c1[i];
      }
    }
  }
  __syncthreads();

  for (int j = tid; j < ROWS_PER_BLOCK; j += THREADS)
    if (blockStart + j >= n_items) sc[j] = NEG_INF;
  __syncthreads();

  int k = size_p[0];
  if (k < 1) k = 1;
  if (k > KMAX) k = KMAX;
  if (k > n_items) k = n_items;

  for (int t = 0; t < k; ++t) {
    float best = NEG_INF;
    int bi = 0x7FFFFFFF;
    for (int j = tid; j < ROWS_PER_BLOCK; j += THREADS) {
      float s = sc[j];
      if (s > best || (s == best && j < bi)) { best = s; bi = j; }
    }
    red_s[tid] = best; red_i[tid] = bi;
    __syncthreads();
    for (int off = THREADS / 2; off > 0; off >>= 1) {
      if (tid < off) {
        float s2 = red_s[tid + off]; int i2 = red_i[tid + off];
        if (s2 > red_s[tid] || (s2 == red_s[tid] && i2 < red_i[tid])) {
          red_s[tid] = s2; red_i[tid] = i2;
        }
      }
      __syncthreads();
    }
    if (tid == 0) {
      ws_s[blockIdx.x * KMAX + t] = red_s[0];
      ws_i[blockIdx.x * KMAX + t] = blockStart + red_i[0];
      sc[red_i[0]] = NEG_INF;
    }
    __syncthreads();
  }
  // pad unused candidate slots so phase 2 can scan a fixed stride
  if (tid < KMAX - k) {
    ws_s[blockIdx.x * KMAX + k + tid] = NEG_INF;
    ws_i[blockIdx.x * KMAX + k + tid] = 0;
  }
}

// Phase 2: single block reduces nblocks*KMAX candidates to global top-k
__global__ void __launch_bounds__(THREADS)
final_topk(float* __restrict__ ws_s, const int* __restrict__ ws_i,
           const int* __restrict__ size_p, float* __restrict__ out,
           int n_items, int nblocks) {
  __shared__ float red_s[THREADS];
  __shared__ int   red_r[THREADS];
  __shared__ int   red_j[THREADS];
  const int tid = threadIdx.x;
  const float NEG_INF = -__builtin_inff();
  const int NC = nblocks * KMAX;

  int k = size_p[0];
  if (k < 1) k = 1;
  if (k > KMAX) k = KMAX;
  if (k > n_items) k = n_items;

  for (int t = 0; t < k; ++t) {
    float best = NEG_INF;
    int brow = 0x7FFFFFFF;
    int bj = -1;
    for (int j = tid; j < NC; j += THREADS) {
      float s = ws_s[j];
      int r = ws_i[j];
      if (s > best || (s == best && r < brow)) { best = s; brow = r; bj = j; }
    }
    red_s[tid] = best; red_r[tid] = brow; red_j[tid] = bj;
    __syncthreads();
    for (int off = THREADS / 2; off > 0; off >>= 1) {
      if (tid < off) {
        if (red_s[tid + off] > red_s[tid] ||
            (red_s[tid + off] == red_s[tid] && red_r[tid + off] < red_r[tid])) {
          red_s[tid] = red_s[tid + off];
          red_r[tid] = red_r[tid + off];
          red_j[tid] = red_j[tid + off];
        }
      }
      __syncthreads();
    }
    if (tid == 0) {
      out[t]     = red_s[0];
      out[k + t] = (float)red_r[0];
      if (red_j[0] >= 0) ws_s[red_j[0]] = NEG_INF;
    }
    __syncthreads();
  }
}

extern "C" void kernel_launch(void* const* d_in, const int* in_sizes, int n_in,
                              void* d_out, int out_size, void* d_ws, size_t ws_size,
                              hipStream_t stream) {
  (void)n_in; (void)out_size; (void)ws_size;
  const float* A      = (const float*)d_in[0];
  const int*   item_p = (const int*)d_in[1];
  const int*   size_p = (const int*)d_in[2];
  float*       out    = (float*)d_out;

  const int n_items = in_sizes[0] / NF;
  const int nblocks = (n_items + ROWS_PER_BLOCK - 1) / ROWS_PER_BLOCK;

  float* ws_s = (float*)d_ws;
  int*   ws_i = (int*)((char*)d_ws + (size_t)nblocks * KMAX * sizeof(float));

  gemv_topk_block<<<nblocks, THREADS, 0, stream>>>(A, item_p, size_p, ws_s, ws_i, n_items);
  final_topk<<<1, THREADS, 0, stream>>>(ws_s, ws_i, size_p, out, n_items, nblocks);
}